// Model_67499706024326
// MI455X (gfx1250) — compile-verified
//
#include <hip/hip_runtime.h>
#include <math.h>

typedef float v2f __attribute__((ext_vector_type(2)));
typedef float v8f __attribute__((ext_vector_type(8)));
typedef unsigned int u32x4 __attribute__((ext_vector_type(4)));
typedef int i32x4 __attribute__((ext_vector_type(4)));
typedef int i32x8 __attribute__((ext_vector_type(8)));

#define TT 256          // signal length
#define CC 64           // channels
#define NFEAT 904       // features per channel
#define FSTR (NFEAT*CC) // 57856, W row stride
#define XS 72           // LDS stride of staged x rows (64 + 8 pad, TDM-inserted)
#define MS 66           // LDS stride of |X| rows (8*66 % 64 == 16 -> halves disjoint)
#define YS 72           // LDS stride of Y rows (2*72 % 64 == 16 -> B-loads disjoint)
#define YROWS 132       // Hilbert-masked spectrum rows (129 used, pad to K-mult of 4)

// D = A(16x4 f32) * B(4x16 f32) + C(16x16 f32), full-precision WMMA
__device__ __forceinline__ v8f wmma4(v2f a, v2f b, v8f c) {
  return __builtin_amdgcn_wmma_f32_16x16x4_f32(false, a, false, b, (short)0, c,
                                               false, false);
}

__global__ __launch_bounds__(256) void spectral_fused(
    const float* __restrict__ x, const float* __restrict__ W,
    const float* __restrict__ bias, float* __restrict__ out)
{
  extern __shared__ float sm[];
  float* sX    = sm;                 // 256*XS  staged x (TDM, padded stride 72)
  float* sTab  = sX + TT*XS;         // 256     cos(2*pi*j/256)
  float* sMag  = sTab + 256;         // 256*MS  |X|
  float* sYr   = sMag + 256*MS;      // YROWS*YS  (h*X/256).re
  float* sYi   = sYr + YROWS*YS;     // YROWS*YS  (h*X/256).im
  float* sMean = sYi + YROWS*YS;     // 64  sum_t x
  float* sSmag = sMean + 64;         // 64  sum mag
  float* sSkm  = sSmag + 64;         // 64  sum f*mag
  float* sS2   = sSkm + 64;          // 64  sum mag^2
  float* sCos  = sS2 + 64;           // 64  sum cos(phase)
  float* sSin  = sCos + 64;          // 64  sum sin(phase)
  float* sSe   = sSin + 64;          // 64  sum |analytic|
  float* sSe2  = sSe + 64;           // 64  sum |analytic|^2
  int*   sPkI  = (int*)(sSe2 + 64);  // 64  peak (float bits, mag>=0)
  float* sBdot = (float*)(sPkI + 64);// 5*64 bicoh dot with W (pre-normalization)
  float* sOut  = sBdot + 5*64;       // 8   output accumulators

  const int tid  = threadIdx.x;
  const int lane = tid & 31;
  const int wave = tid >> 5;
  const int half = lane >> 4;   // which 16-lane half
  const int l16  = lane & 15;
  const int bb   = blockIdx.x;
  const float* xb = x + (size_t)bb * (TT * CC);

  // -------- TDM: stage x_b (16384 f32) into LDS with pad-stride 72 --------
  if (wave == 0) {
    const unsigned long long ga = (unsigned long long)(uintptr_t)xb;
    const unsigned ldsOff = (unsigned)(uintptr_t)sX;  // low 32 bits = LDS offset
    u32x4 g0;
    g0[0] = 1u;                                   // count=1, no gather
    g0[1] = ldsOff;                               // lds_addr (bytes)
    g0[2] = (unsigned)(ga & 0xFFFFFFFFu);         // global_addr[31:0]
    g0[3] = (unsigned)((ga >> 32) & 0x01FFFFFFu)  // global_addr[56:32]
          | 0x80000000u;                          // type=2 ("image")
    i32x8 g1;
    // data_size=2(4B) | pad_enable | pad_interval=5(64 dw) | pad_amount=7(8 dw)
    g1[0] = (int)((2u << 16) | (1u << 20) | (5u << 22) | (7u << 25));
    g1[1] = (int)(16384u << 16);                  // tensor_dim0[15:0]=16384
    g1[2] = (int)(1u << 16);                      // tensor_dim0 hi=0, tensor_dim1=1
    g1[3] = (int)(16384u << 16);                  // tensor_dim1 hi=0, tile_dim0=16384
    g1[4] = 0;                                    // tile_dim1=0, tile_dim2=0
    g1[5] = 16384;                                // tensor_dim0_stride[31:0]
    g1[6] = 0; g1[7] = 0;
    i32x4 gz4 = {0, 0, 0, 0};
    i32x8 gz8 = {0, 0, 0, 0, 0, 0, 0, 0};
    __builtin_amdgcn_tensor_load_to_lds(g0, g1, gz4, gz4, gz8, 0);
  }

  // ---------------- init (overlaps with TDM) ----------------
  sTab[tid] = cosf((float)(2.0 * M_PI / 256.0) * (float)tid);
  for (int i = tid; i < 904; i += 256) sMean[i] = 0.0f;      // stats/bdot/out
  for (int i = tid; i < 3 * YS; i += 256) {                  // Y pad rows 129..131
    sYr[129 * YS + i] = 0.0f; sYi[129 * YS + i] = 0.0f;
  }
  __builtin_prefetch(W + (size_t)tid * 904, 0, 1);           // warm W into L2

  if (wave == 0) __builtin_amdgcn_s_wait_tensorcnt(0);       // TDM done (wave 0)
  __syncthreads();                                           // publish sX to WG

  // ---------------- per-channel mean of x (from LDS) ----------------
  {
    int c = tid & 63, part = tid >> 6;
    float s = 0.0f;
    for (int t = 0; t < 64; ++t) s += sX[(part * 64 + t) * XS + c];
    atomicAdd(&sMean[c], s);
  }
  __syncthreads();

  float pout[5] = {0.f, 0.f, 0.f, 0.f, 0.f};  // register partials of out[b,:]

  // ============ GEMM1: X = F @ x  (M=256 freq, N=64 ch, K=256 time) ============
  for (int tile = wave; tile < 64; tile += 8) {
    const int mt = tile >> 2, nt = tile & 3;
    const int c  = nt * 16 + l16;
    const int m  = mt * 16 + l16;  // A row (freq)
    v8f accR = {}; v8f accI = {};
    for (int kk = 0; kk < 64; ++kk) {
      const int t0 = kk * 4 + half * 2;         // K element index
      v2f Bv = { sX[t0 * XS + c], sX[(t0 + 1) * XS + c] };
      unsigned j0 = (unsigned)(m * t0) & 255u;
      unsigned j1 = (unsigned)(m * (t0 + 1)) & 255u;
      v2f Ar = { sTab[j0], sTab[j1] };                               //  cos
      v2f Ai = { sTab[(j0 + 64) & 255u], sTab[(j1 + 64) & 255u] };   // -sin
      accR = wmma4(Ar, Bv, accR);
      accI = wmma4(Ai, Bv, accI);
    }
    float lS = 0.f, lK = 0.f, lS2 = 0.f, lC = 0.f, lSi = 0.f, lPk = 0.f;
    float bl[5] = {0.f, 0.f, 0.f, 0.f, 0.f};
#pragma unroll
    for (int v = 0; v < 8; ++v) {
      const int k = mt * 16 + v + half * 8;     // C/D row = vgpr + 8*(lane>=16)
      const float r = accR[v], im = accI[v];
      const float mg = sqrtf(r * r + im * im);
      sMag[k * MS + c] = mg;
      const float ph = atan2f(im, r);
      const float fk = (k < 128) ? (float)k : (float)(k - 256);
      lS += mg; lK += fk * mg; lS2 += mg * mg;
      if (mg > 0.f) { float inv = 1.0f / mg; lC += r * inv; lSi += im * inv; }
      else          { lC += 1.0f; }
      lPk = fmaxf(lPk, mg);
      if (k <= 128) {
        const float h = (k == 0 || k == 128) ? (1.0f / 256.0f) : (2.0f / 256.0f);
        sYr[k * YS + c] = r * h; sYi[k * YS + c] = im * h;
      }
#pragma unroll
      for (int n = 0; n < 5; ++n) {
        const float* Wn = W + n * FSTR;
        pout[n] += ph * Wn[k * CC + c] + mg * Wn[(257 + k) * CC + c];
        bl[n]   += mg * Wn[(513 + k) * CC + c];
      }
    }
    atomicAdd(&sSmag[c], lS);  atomicAdd(&sSkm[c], lK);
    atomicAdd(&sS2[c], lS2);   atomicAdd(&sCos[c], lC);
    atomicAdd(&sSin[c], lSi);  atomicMax(&sPkI[c], __float_as_int(lPk));
#pragma unroll
    for (int n = 0; n < 5; ++n) atomicAdd(&sBdot[n * 64 + c], bl[n]);
  }

  // ==== GEMM2: Xw = (F .* win) @ x - mean*HannDFT  (rows 0..128 used) ====
  for (int tile = wave; tile < 36; tile += 8) {
    const int mt = tile >> 2, nt = tile & 3;
    const int c  = nt * 16 + l16;
    const int m  = mt * 16 + l16;
    v8f accR = {}; v8f accI = {};
    for (int kk = 0; kk < 64; ++kk) {
      const int t0 = kk * 4 + half * 2;
      v2f Bv = { sX[t0 * XS + c], sX[(t0 + 1) * XS + c] };
      const float w0 = 0.5f - 0.5f * sTab[t0 & 255];
      const float w1 = 0.5f - 0.5f * sTab[(t0 + 1) & 255];
      unsigned j0 = (unsigned)(m * t0) & 255u;
      unsigned j1 = (unsigned)(m * (t0 + 1)) & 255u;
      v2f Ar = { sTab[j0] * w0, sTab[j1] * w1 };
      v2f Ai = { sTab[(j0 + 64) & 255u] * w0, sTab[(j1 + 64) & 255u] * w1 };
      accR = wmma4(Ar, Bv, accR);
      accI = wmma4(Ai, Bv, accI);
    }
    const float msum = sMean[c];   // sum_t x ; mean = msum/256
#pragma unroll
    for (int v = 0; v < 8; ++v) {
      const int k = mt * 16 + v + half * 8;
      if (k > 128) continue;
      float Xwr = accR[v], Xwi = accI[v];
      // Hann DFT: Wsum(0)=128, Wsum(1)=-64 (real); Xw -= mean*Wsum
      if (k == 0)      Xwr -= msum * 0.5f;
      else if (k == 1) Xwr += msum * 0.25f;
      const float mult = (k == 0 || k == 128) ? 1.0f : 2.0f;
      const float psd = (Xwr * Xwr + Xwi * Xwi) * (1.0f / 24576.0f) * mult;
#pragma unroll
      for (int n = 0; n < 5; ++n)
        pout[n] += psd * W[n * FSTR + (773 + k) * CC + c];
    }
  }
  __syncthreads();   // sYr/sYi complete before inverse pass

  // ==== GEMM3: analytic = conj(F)/T @ (h.*X)  (M=256 time, K=132) ====
  for (int tile = wave; tile < 64; tile += 8) {
    const int mt = tile >> 2, nt = tile & 3;
    const int c  = nt * 16 + l16;
    const int t  = mt * 16 + l16;   // A row (time)
    v8f aR = {}; v8f aI = {};
    for (int kk = 0; kk < 33; ++kk) {
      const int k0 = kk * 4 + half * 2;
      unsigned j0 = (unsigned)(t * k0) & 255u;
      unsigned j1 = (unsigned)(t * (k0 + 1)) & 255u;
      v2f Ac = { sTab[j0], sTab[j1] };                                 //  cos
      v2f Am = { sTab[(j0 + 64) & 255u],  sTab[(j1 + 64) & 255u] };    // -sin
      v2f As = { sTab[(j0 + 192) & 255u], sTab[(j1 + 192) & 255u] };   // +sin
      v2f Br = { sYr[k0 * YS + c], sYr[(k0 + 1) * YS + c] };
      v2f Bi = { sYi[k0 * YS + c], sYi[(k0 + 1) * YS + c] };
      aR = wmma4(Ac, Br, aR);  aR = wmma4(Am, Bi, aR);  // re = cos*Yr - sin*Yi
      aI = wmma4(Ac, Bi, aI);  aI = wmma4(As, Br, aI);  // im = cos*Yi + sin*Yr
    }
    float lE = 0.f, lE2 = 0.f;
#pragma unroll
    for (int v = 0; v < 8; ++v) {
      const float env = sqrtf(aR[v] * aR[v] + aI[v] * aI[v]);
      lE += env; lE2 += env * env;
    }
    atomicAdd(&sSe[c], lE); atomicAdd(&sSe2[c], lE2);
  }

#pragma unroll
  for (int n = 0; n < 5; ++n) atomicAdd(&sOut[n], pout[n]);
  __syncthreads();

  // ---------------- per-channel scalar features ----------------
  if (tid < 64) {
    const int c = tid;
    const float Smag = sSmag[c];
    const float pcoh = sqrtf(sCos[c] * sCos[c] + sSin[c] * sSin[c]) * (1.0f / 256.0f);
    const float centroid = sSkm[c] / Smag;
    const float avg  = Smag * (1.0f / 256.0f);
    const float peak = __int_as_float(sPkI[c]);
    const float bs   = rsqrtf(sS2[c]);
    const float m1   = sSe[c] * (1.0f / 256.0f);
    const float amp  = sqrtf(fmaxf(sSe2[c] * (1.0f / 256.0f) - m1 * m1, 0.0f));
    // cumsum scan for rolloff / median frequency
    float csum = 0.f; const float thr1 = 0.85f * Smag, thr2 = 0.5f * Smag;
    int i1 = -1, i2 = -1;
    for (int k = 0; k < 256; ++k) {
      csum += sMag[k * MS + c];
      if (i1 < 0 && csum >= thr1) i1 = k;
      if (i2 < 0 && csum >= thr2) i2 = k;
    }
    if (i1 < 0) i1 = 255; if (i2 < 0) i2 = 255;
    const float roll = (float)i1;
    const float medf = (i2 < 128) ? (float)i2 : (float)(i2 - 256);
#pragma unroll
    for (int n = 0; n < 5; ++n) {
      const float* Wn = W + n * FSTR;
      float s = pcoh     * Wn[256 * CC + c]
              + centroid * Wn[769 * CC + c]
              + roll     * Wn[770 * CC + c]
              + peak     * Wn[771 * CC + c]
              + avg      * Wn[772 * CC + c]
              + medf     * Wn[902 * CC + c]
              + amp      * Wn[903 * CC + c]
              + bs       * sBdot[n * 64 + c];
      atomicAdd(&sOut[n], s);
    }
  }
  __syncthreads();

  if (tid < 5) out[bb * 5 + tid] = sOut[tid] + bias[tid];
}

extern "C" void kernel_launch(void* const* d_in, const int* in_sizes, int n_in,
                              void* d_out, int out_size, void* d_ws, size_t ws_size,
                              hipStream_t stream) {
  (void)n_in; (void)out_size; (void)d_ws; (void)ws_size;
  const float* x  = (const float*)d_in[0];
  // d_in[1] (x_mark_enc) is unused by the reference computation
  const float* W  = (const float*)d_in[2];
  const float* bv = (const float*)d_in[3];
  float* out = (float*)d_out;
  const int Bn = in_sizes[0] / (TT * CC);   // 1024
  const size_t smem =
      (size_t)(TT * XS + 256 + 256 * MS + 2 * YROWS * YS + 904) * sizeof(float);
  (void)hipFuncSetAttribute((const void*)spectral_fused,
                            hipFuncAttributeMaxDynamicSharedMemorySize, (int)smem);
  spectral_fused<<<dim3(Bn), dim3(256), smem, stream>>>(x, W, bv, out);
}